// GCNEncoder_31147102831240
// MI455X (gfx1250) — compile-verified
//
#include <hip/hip_runtime.h>
#include <math.h>

// ---------------------------------------------------------------------------
// GCN encoder for MI455X (gfx1250, wave32).
//   h    = gcn_conv(x, W1, b1)            [N,64]
//   h    = relu(layer_norm(h))            [N,64]
//   out1 = gcn_conv(h, W2, b2)            [N,64]
//   out2 = sigmoid(h @ W3 + b3)           [N,6]
// Dense GEMMs use V_WMMA_F32_16X16X4_F32 (fp32 matrix pipe); W is staged in
// LDS pre-swizzled into B-fragment layout so each fragment is one ds_load_b64
// with a conflict-free bank pattern. Edge aggregation uses relaxed agent-scope
// global_atomic_add_f32 into L2-resident buffers (192MB L2 >> 12.8MB feature
// buffer), which is the workload's true cost center — the GEMMs are ~0.6 GMAC.
// ---------------------------------------------------------------------------

typedef __attribute__((ext_vector_type(2))) float v2f;
typedef __attribute__((ext_vector_type(8))) float v8f;

#define OUTC 64

__device__ __forceinline__ void atomicAddF32(float* p, float v) {
    __hip_atomic_fetch_add(p, v, __ATOMIC_RELAXED, __HIP_MEMORY_SCOPE_AGENT);
}

// ---------------- degree / dinv ----------------
__global__ void k_fill1(float* __restrict__ d, int n) {
    int i = blockIdx.x * blockDim.x + threadIdx.x;
    if (i < n) d[i] = 1.0f;                      // self-loop contribution
}

__global__ void k_deg_edges(const int* __restrict__ col, float* __restrict__ deg, int E) {
    int e = blockIdx.x * blockDim.x + threadIdx.x;
    if (e < E) atomicAddF32(&deg[col[e]], 1.0f);
}

__global__ void k_to_dinv(float* __restrict__ d, int n) {
    int i = blockIdx.x * blockDim.x + threadIdx.x;
    if (i < n) d[i] = rsqrtf(d[i]);              // deg >= 1 always
}

// ---------------- WMMA GEMM: C[M x 64] = A[M x K] * W[K x 64] ----------------
// One wave computes a 16x64 tile: 4 accumulators of 16x16 f32, K-loop step 4.
// A-frag (16x4 f32, 2 VGPR): lanes 0-15 hold M=lane,    K = k0 + {0,1};
//                            lanes 16-31 hold M=lane-16, K = k0 + {2,3}.
// B-frag (4x16 f32, 2 VGPR): mirrored; per lane it needs the pair
//   ( W[k0+2*half][n], W[k0+2*half+1][n] )  with n = ntile*16 + (lane&15).
// We stage W in LDS as  ldsB[kg][n][half] = that float2  (kg = k0/4), so a
// fragment is ONE ds_load_b64 and lane->bank is n*4 + half*2: lanes 0-15 hit
// banks {0,1,4,5,...}, lanes 16-31 hit {2,3,6,7,...} -> conflict-free.
// C/D (8 VGPR): element r of lane l is (M = r + 8*(l>>4), N = l&15).
__global__ void __launch_bounds__(256)
k_gemm_wmma(const float* __restrict__ A, const float* __restrict__ W,
            float* __restrict__ C, int M, int K) {
    __shared__ v2f ldsB[32 * 128];               // (K/4)*64*2 float2, <=32KB
    const int tid = threadIdx.x;
    for (int idx = tid; idx < K * 32; idx += 256) {
        const int kg   = idx >> 7;               // k-group of 4
        const int r    = idx & 127;
        const int n    = r >> 1;
        const int half = r & 1;
        const int kk   = kg * 4 + 2 * half;
        v2f v;
        v.x = W[kk * OUTC + n];
        v.y = W[kk * OUTC + OUTC + n];
        ldsB[idx] = v;
    }
    __syncthreads();

    const int wave = tid >> 5;
    const int lane = tid & 31;
    const int half = lane >> 4;                  // 0: lanes 0-15, 1: lanes 16-31
    const int mr   = lane & 15;
    const int rowBase = (blockIdx.x * 8 + wave) * 16;
    if (rowBase >= M) return;                    // wave-uniform -> EXEC stays all-1s

    v8f acc0 = {}, acc1 = {}, acc2 = {}, acc3 = {};
    const float* Arow = A + (size_t)(rowBase + mr) * K;

    const int KG = K >> 2;
    for (int kg = 0; kg < KG; ++kg) {
        const int kk = kg * 4 + 2 * half;
        v2f a;
        a.x = Arow[kk];
        a.y = Arow[kk + 1];

        const v2f* bq = &ldsB[kg * 128 + mr * 2 + half];
        const v2f b0 = bq[0];                    // ntile 0: n = mr
        const v2f b1 = bq[32];                   // ntile 1: n = mr + 16
        const v2f b2 = bq[64];                   // ntile 2: n = mr + 32
        const v2f b3 = bq[96];                   // ntile 3: n = mr + 48

        acc0 = __builtin_amdgcn_wmma_f32_16x16x4_f32(false, a, false, b0, (short)0, acc0, false, false);
        acc1 = __builtin_amdgcn_wmma_f32_16x16x4_f32(false, a, false, b1, (short)0, acc1, false, false);
        acc2 = __builtin_amdgcn_wmma_f32_16x16x4_f32(false, a, false, b2, (short)0, acc2, false, false);
        acc3 = __builtin_amdgcn_wmma_f32_16x16x4_f32(false, a, false, b3, (short)0, acc3, false, false);
    }

    float* Cw = C + (size_t)(rowBase + 8 * half) * OUTC + mr;
#pragma unroll
    for (int r = 0; r < 8; ++r) {
        Cw[(size_t)r * OUTC +  0] = acc0[r];
        Cw[(size_t)r * OUTC + 16] = acc1[r];
        Cw[(size_t)r * OUTC + 32] = acc2[r];
        Cw[(size_t)r * OUTC + 48] = acc3[r];
    }
}

// ---------------- aggregation ----------------
// dst[i] = dinv[i]^2 * src[i] + bias  (self-loop term + conv bias)
__global__ void k_agg_init(const float* __restrict__ dinv, const float* __restrict__ src,
                           const float* __restrict__ bias, float* __restrict__ dst, int n) {
    int g = blockIdx.x * blockDim.x + threadIdx.x;
    int node = g >> 5, lane = g & 31;
    if (node >= n) return;
    float s = dinv[node]; s *= s;
    float2 v = ((const float2*)(src + (size_t)node * OUTC))[lane];
    float2 o;
    o.x = s * v.x + bias[2 * lane + 0];
    o.y = s * v.y + bias[2 * lane + 1];
    ((float2*)(dst + (size_t)node * OUTC))[lane] = o;
}

// dst[col[e]] += dinv[row]*dinv[col] * src[row]; one 32-lane group per edge.
__global__ void k_agg_edges(const int* __restrict__ row, const int* __restrict__ col,
                            const float* __restrict__ dinv, const float* __restrict__ src,
                            float* __restrict__ dst, int E) {
    int g = blockIdx.x * blockDim.x + threadIdx.x;
    int e = g >> 5, lane = g & 31;
    if (e >= E) return;
    int r = row[e], c = col[e];
    float nrm = dinv[r] * dinv[c];
    float2 v = ((const float2*)(src + (size_t)r * OUTC))[lane];
    float* d = dst + (size_t)c * OUTC + 2 * lane;
    atomicAddF32(d + 0, nrm * v.x);
    atomicAddF32(d + 1, nrm * v.y);
}

// ---------------- layernorm + relu (one wave per node, wave32 butterflies) ----
__global__ void __launch_bounds__(256)
k_ln_relu(const float* __restrict__ in, const float* __restrict__ w,
          const float* __restrict__ b, float* __restrict__ out, int n) {
    const int tid  = threadIdx.x;
    const int lane = tid & 31;
    const int node = blockIdx.x * 8 + (tid >> 5);
    if (node >= n) return;
    float2 v = ((const float2*)(in + (size_t)node * OUTC))[lane];

    float s = v.x + v.y;
    for (int m = 16; m; m >>= 1) s += __shfl_xor(s, m, 32);
    float mu = s * (1.0f / 64.0f);

    float dx = v.x - mu, dy = v.y - mu;
    float q = dx * dx + dy * dy;
    for (int m = 16; m; m >>= 1) q += __shfl_xor(q, m, 32);
    float rstd = rsqrtf(q * (1.0f / 64.0f) + 1e-5f);

    float2 o;
    o.x = fmaxf(dx * rstd * w[2 * lane + 0] + b[2 * lane + 0], 0.0f);
    o.y = fmaxf(dy * rstd * w[2 * lane + 1] + b[2 * lane + 1], 0.0f);
    ((float2*)(out + (size_t)node * OUTC))[lane] = o;
}

// ---------------- sigmoid head: out2 = sigmoid(h @ W3 + b3), W3 is 64x6 -------
__global__ void k_head(const float* __restrict__ h, const float* __restrict__ W3,
                       const float* __restrict__ b3, float* __restrict__ out, int n) {
    int node = blockIdx.x * blockDim.x + threadIdx.x;
    if (node >= n) return;
    float acc[6];
#pragma unroll
    for (int j = 0; j < 6; ++j) acc[j] = b3[j];
    const float* hr = h + (size_t)node * OUTC;
    for (int k = 0; k < OUTC; ++k) {
        float hv = hr[k];
#pragma unroll
        for (int j = 0; j < 6; ++j) acc[j] += hv * W3[k * 6 + j];
    }
    float* o = out + (size_t)node * 6;
#pragma unroll
    for (int j = 0; j < 6; ++j) o[j] = 1.0f / (1.0f + __expf(-acc[j]));
}

// ---------------------------------------------------------------------------
extern "C" void kernel_launch(void* const* d_in, const int* in_sizes, int n_in,
                              void* d_out, int out_size, void* d_ws, size_t ws_size,
                              hipStream_t stream) {
    const float* x   = (const float*)d_in[0];
    const int*   ei  = (const int*)d_in[1];     // [2, E] flat: rows then cols
    const float* W1  = (const float*)d_in[2];
    const float* b1  = (const float*)d_in[3];
    const float* lnw = (const float*)d_in[4];
    const float* lnb = (const float*)d_in[5];
    const float* W2  = (const float*)d_in[6];
    const float* b2  = (const float*)d_in[7];
    const float* W3  = (const float*)d_in[8];
    const float* b3  = (const float*)d_in[9];

    const int N = in_sizes[0] / 128;            // 50000
    const int E = in_sizes[1] / 2;              // 800000
    const int* row = ei;
    const int* col = ei + E;

    float* out1 = (float*)d_out;                // [N,64]
    float* out2 = out1 + (size_t)N * OUTC;      // [N,6]

    // workspace: dinv[N] | bufA[N*64] | bufB[N*64]   (~25.8 MB)
    float* dinv = (float*)d_ws;
    float* bufA = dinv + (((size_t)N + 63) & ~(size_t)63);
    float* bufB = bufA + (size_t)N * OUTC;

    const int nbN   = (N + 255) / 256;
    const int nbE   = (E + 255) / 256;
    const int nbN32 = (int)(((size_t)N * 32 + 255) / 256);   // 32 lanes per node
    const int nbE32 = (int)(((size_t)E * 32 + 255) / 256);   // 32 lanes per edge
    const int nbGem = (N + 127) / 128;                       // 8 waves x 16 rows

    // degrees -> dinv
    k_fill1    <<<nbN, 256, 0, stream>>>(dinv, N);
    k_deg_edges<<<nbE, 256, 0, stream>>>(col, dinv, E);
    k_to_dinv  <<<nbN, 256, 0, stream>>>(dinv, N);

    // layer 1: gemm + normalized aggregation + LN/ReLU
    k_gemm_wmma<<<nbGem, 256, 0, stream>>>(x, W1, bufA, N, 128);
    k_agg_init <<<nbN32, 256, 0, stream>>>(dinv, bufA, b1, bufB, N);
    k_agg_edges<<<nbE32, 256, 0, stream>>>(row, col, dinv, bufA, bufB, E);
    k_ln_relu  <<<(N + 7) / 8, 256, 0, stream>>>(bufB, lnw, lnb, bufA, N);

    // layer 2: gemm + aggregation straight into out1
    k_gemm_wmma<<<nbGem, 256, 0, stream>>>(bufA, W2, bufB, N, 64);
    k_agg_init <<<nbN32, 256, 0, stream>>>(dinv, bufB, b2, out1, N);
    k_agg_edges<<<nbE32, 256, 0, stream>>>(row, col, dinv, bufB, out1, E);

    // sigmoid head
    k_head<<<nbN, 256, 0, stream>>>(bufA, W3, b3, out2, N);
}